// DictionaryLearning_34419867910813
// MI455X (gfx1250) — compile-verified
//
#include <hip/hip_runtime.h>
#include <math.h>

typedef __attribute__((ext_vector_type(2))) float v2f;
typedef __attribute__((ext_vector_type(4))) float v4f;
typedef __attribute__((ext_vector_type(8))) float v8f;

#define EMBED   64
#define NATOMS  512
#define NCOLS   131072          // 32*64*64
#define NSPARSE 5
#define TBCOLS  128             // columns per block
#define DLEPS   1e-10f

// A-fragment LDS layout for V_WMMA_F32_16X16X4_F32:
//   element (k, n) -> (((at*16 + k/4)*2 + (k/2)%2)*16 + n%16)*2 + k%2
// so a lane's (K=k0,K=k0+1) pair is one aligned 8-byte ds_load_b64 and the
// 16 lanes of each half-wave hit distinct LDS banks (stride-2 / +32 sets).
__device__ __forceinline__ int afIndex(int k, int n) {
    return ((((n >> 4) * 16 + (k >> 2)) * 2 + ((k >> 1) & 1)) * 16 + (n & 15)) * 2 + (k & 1);
}

// ---------------------------------------------------------------------------
// Kernel 1: column-normalize dictionary, record den[n] = sum(dn^2) (~1.0)
// ---------------------------------------------------------------------------
__global__ __launch_bounds__(512) void dl_normalize(const float* __restrict__ dict,
                                                    float* __restrict__ dn,
                                                    float* __restrict__ den) {
    int n = threadIdx.x;                       // one thread per atom
    float s = 0.f;
    for (int m = 0; m < EMBED; ++m) { float v = dict[m*NATOMS + n]; s += v*v; }
    float inv = 1.0f / sqrtf(s);
    float d2 = 0.f;
    for (int m = 0; m < EMBED; ++m) {
        float v = dict[m*NATOMS + n] * inv;
        dn[m*NATOMS + n] = v;
        d2 += v*v;
    }
    den[n] = d2;
}

// ---------------------------------------------------------------------------
// Kernel 2: zero loss+coeffs region with 16B stores (region is 16B aligned)
// ---------------------------------------------------------------------------
__global__ void dl_zero4(float* __restrict__ base, unsigned long long nvec) {
    unsigned long long i = (unsigned long long)blockIdx.x * blockDim.x + threadIdx.x;
    unsigned long long stride = (unsigned long long)gridDim.x * blockDim.x;
    v4f z = {0.f, 0.f, 0.f, 0.f};
    v4f* p = (v4f*)base;
    for (; i < nvec; i += stride) p[i] = z;
    if (blockIdx.x == 0 && threadIdx.x == 0) base[nvec * 4] = 0.f;  // tail float
}

// ---------------------------------------------------------------------------
// Kernel 3: fused 5-round batch-OMP, 128 columns per block.
// Block -> column set: col = h5*4096 + w*64 + (c0 + cs), local j = cs*64 + w,
// with h5 = blockIdx>>5 and c0 = (blockIdx&31)*2. This makes all global z
// traffic coalesced (runs of 64 consecutive floats over w).
// ---------------------------------------------------------------------------
__global__ __launch_bounds__(256) void dl_omp(const float* __restrict__ ze,
                                              const float* __restrict__ dnG,
                                              const float* __restrict__ den,
                                              float* __restrict__ zout,
                                              float* __restrict__ coeffs,
                                              float* __restrict__ blockSums) {
    __shared__ float AF[NATOMS * EMBED];        // 128 KB: Dn^T in A-frag layout
    __shared__ float R[EMBED][TBCOLS];          //  32 KB: residual tile

    const int tid   = threadIdx.x;
    const int lane  = tid & 31;
    const int wave  = tid >> 5;                 // 8 waves, 16 columns each
    const int n0    = lane & 15;
    const int khalf = lane >> 4;
    const int c0    = (blockIdx.x & 31) << 1;
    const int h5    = blockIdx.x >> 5;

    // Stage residual tile (coalesced: 64-float runs over w)
    for (int i = tid; i < EMBED * TBCOLS; i += 256) {
        int m = i >> 7, j = i & (TBCOLS - 1);
        int w = j & 63, cs = j >> 6;
        int b = m >> 1, h = ((m & 1) << 5) + h5;
        int g = ((b * 64 + c0 + cs) * 64 + h) * 64 + w;
        R[m][j] = ze[g];
    }
    // Stage Dn^T into LDS in A-fragment order (coalesced global reads)
    for (int i = tid; i < EMBED * NATOMS; i += 256) {
        int k = i >> 9, n = i & (NATOMS - 1);
        AF[afIndex(k, n)] = dnG[i];
    }
    __syncthreads();

    const int wcol = wave * 16 + n0;            // this lane's local column
    int   selIdx[NSPARSE];
    float selAlpha[NSPARSE];

    for (int it = 0; it < NSPARSE; ++it) {
        // B fragments for this column (live in registers across atom tiles)
        v2f bfr[16];
#pragma unroll
        for (int t = 0; t < 16; ++t) {
            int k0 = t * 4 + khalf * 2;
            bfr[t].x = R[k0][wcol];
            bfr[t].y = R[k0 + 1][wcol];
        }
        // branchless argmax key: |corr|<<32 | (511-idx)<<1 | sign
        unsigned long long bestKey = 0ull;
        for (int at = 0; at < NATOMS / 16; ++at) {
            v8f acc = {};
            const int atomBase = at << 4;
#pragma unroll
            for (int t = 0; t < 16; ++t) {
                int k0 = t * 4 + khalf * 2;
                v2f a = *(const v2f*)&AF[afIndex(k0, atomBase + n0)];
                acc = __builtin_amdgcn_wmma_f32_16x16x4_f32(
                          false, a, false, bfr[t], (short)0, acc, false, false);
            }
#pragma unroll
            for (int v = 0; v < 8; ++v) {
                float val = acc[v];
                unsigned int ab = __float_as_uint(fabsf(val));
                int idx = atomBase + v + (khalf << 3);
                unsigned long long key =
                    ((unsigned long long)ab << 32) |
                    ((unsigned long long)(unsigned)(511 - idx) << 1) |
                    (unsigned long long)(val < 0.f ? 1u : 0u);
                bestKey = key > bestKey ? key : bestKey;
            }
        }
        // merge the two half-wave candidates for the same column
        unsigned long long oKey = __shfl_xor(bestKey, 16);
        bestKey = oKey > bestKey ? oKey : bestKey;

        int bestIdx   = 511 - (int)((bestKey >> 1) & 511u);
        float av      = __uint_as_float((unsigned int)(bestKey >> 32));
        float bestVal = (bestKey & 1ull) ? -av : av;
        float alpha   = bestVal / (den[bestIdx] + DLEPS);
        selIdx[it] = bestIdx; selAlpha[it] = alpha;

        // rank-1 residual update, 64 rows split across the lane pair
        const int mBase = khalf << 5;
        for (int m = 0; m < 32; ++m)
            R[mBase + m][wcol] -= AF[afIndex(mBase + m, bestIdx)] * alpha;
        __syncthreads();
    }

    // coefficient scatter-add (unique owner lane per column; pre-zeroed)
    if (khalf == 0) {
        int gc = h5 * 4096 + (wcol & 63) * 64 + c0 + (wcol >> 6);
        for (int it = 0; it < NSPARSE; ++it) {
            float* p = coeffs + (size_t)selIdx[it] * NCOLS + gc;
            *p += selAlpha[it];
        }
    }

    // z_out = X - R (coalesced), accumulate sum(R^2) for the loss
    float part = 0.f;
    for (int i = tid; i < EMBED * TBCOLS; i += 256) {
        int m = i >> 7, j = i & (TBCOLS - 1);
        int w = j & 63, cs = j >> 6;
        int b = m >> 1, h = ((m & 1) << 5) + h5;
        int g = ((b * 64 + c0 + cs) * 64 + h) * 64 + w;
        float r = R[m][j];
        zout[g] = ze[g] - r;
        part += r * r;
    }
    for (int off = 16; off > 0; off >>= 1) part += __shfl_down(part, off);
    __syncthreads();                 // all reads of R are done
    if (lane == 0) R[0][wave] = part;   // reuse dead LDS for block reduction
    __syncthreads();
    if (tid == 0) {
        float s = 0.f;
        for (int w8 = 0; w8 < 8; ++w8) s += R[0][w8];
        blockSums[blockIdx.x] = s;
    }
}

// ---------------------------------------------------------------------------
// Kernel 4: deterministic final reduction. loss = 1.25 * mean(R^2)
// ---------------------------------------------------------------------------
__global__ void dl_loss(const float* __restrict__ blockSums,
                        float* __restrict__ loss, int nblocks) {
    float s = 0.f;
    for (int i = 0; i < nblocks; ++i) s += blockSums[i];
    *loss = 1.25f * s / (float)((size_t)EMBED * NCOLS);
}

// ---------------------------------------------------------------------------
extern "C" void kernel_launch(void* const* d_in, const int* in_sizes, int n_in,
                              void* d_out, int out_size, void* d_ws, size_t ws_size,
                              hipStream_t stream) {
    const float* ze   = (const float*)d_in[0];   // (32,64,64,64) f32
    const float* dict = (const float*)d_in[1];   // (64,512) f32
    float* out = (float*)d_out;

    // d_out layout: z_out (8388608) | loss (1) | coeffs (67108864)
    float* zout   = out;
    float* loss   = out + (size_t)EMBED * NCOLS;
    float* coeffs = loss + 1;

    // d_ws layout: den (512) | Dn (64*512) | blockSums (1024)  ~135 KB
    float* den       = (float*)d_ws;
    float* dn        = den + NATOMS;
    float* blockSums = dn + EMBED * NATOMS;

    const int nblocks = NCOLS / TBCOLS;          // 1024

    dl_normalize<<<1, 512, 0, stream>>>(dict, dn, den);
    // zero loss + coeffs: 67108865 floats = 16777216 float4 + 1 tail
    dl_zero4<<<4096, 256, 0, stream>>>(loss, 16777216ull);
    dl_omp<<<nblocks, 256, 0, stream>>>(ze, dn, den, zout, coeffs, blockSums);
    dl_loss<<<1, 1, 0, stream>>>(blockSums, loss, nblocks);
}